// TensplitGCNLARGE_5849745457616
// MI455X (gfx1250) — compile-verified
//
#include <hip/hip_runtime.h>

typedef float v2f __attribute__((ext_vector_type(2)));
typedef float v8f __attribute__((ext_vector_type(8)));

#define IN_DIM 512
#define HIDDEN 128
#define NCLS   64

// ---------------------------------------------------------------------------
// Fused dense phase: H1 = relu(X @ W0) @ W1, via V_WMMA_F32_16X16X4_F32.
// One wave owns two 16-row tiles. A-fragment layout (f32 16x4):
//   lane m = lane&15 is row M; lanes 0-15 carry K=k,k+1 (v0,v1),
//   lanes 16-31 carry K=k+2,k+3  -> per-lane v2f load at k + 2*(lane>>4).
// B-fragment layout (f32 4x16): lane n = lane&15 is col N;
//   v0 = B[kbase][n], v1 = B[kbase+1][n], kbase = 2*(lane>>4).
// C/D layout: VGPR r holds row (r + 8*(lane>>4)), col = lane&15.
// ---------------------------------------------------------------------------
__global__ void __launch_bounds__(256) gcn_dense_kernel(
    const float* __restrict__ X, const float* __restrict__ W0,
    const float* __restrict__ W1, float* __restrict__ H1, int nTiles16)
{
  // 8 waves/block, 8 KB per wave for the C->A transpose of one 16x128 tile.
  __shared__ float hs_all[8][16 * HIDDEN];

  const int wave  = (int)(threadIdx.x >> 5);
  const int lane  = (int)(threadIdx.x & 31);
  const int m     = lane & 15;          // row (A) / col (B,C)
  const int koff  = (lane >> 4) << 1;   // 0 or 2: K sub-offset per lane half
  const int rbase = (lane >> 4) << 3;   // 0 or 8: C/D row base per lane half

  const int tile0 = (blockIdx.x * 8 + wave) * 2;  // two 16-row tiles per wave
  if (tile0 >= nTiles16) return;                  // uniform per wave: EXEC stays all-1
  const bool has2 = (tile0 + 1) < nTiles16;

  float* hs = &hs_all[wave][0];

  v8f acc0[8], acc1[8];
#pragma unroll
  for (int i = 0; i < 8; ++i)
#pragma unroll
    for (int j = 0; j < 8; ++j) { acc0[i][j] = 0.0f; acc1[i][j] = 0.0f; }

  const float* a0p = X + (size_t)(tile0 * 16 + m) * IN_DIM;
  const float* a1p = X + (size_t)((has2 ? (tile0 + 1) : tile0) * 16 + m) * IN_DIM;

  // GEMM1: acc[ct] (16x16 tiles of the 16x128 hidden strip), K = 512.
  for (int k = 0; k < IN_DIM; k += 4) {
    v2f a0 = *(const v2f*)(a0p + k + koff);
    v2f a1 = *(const v2f*)(a1p + k + koff);
    const float* bp = W0 + (size_t)(k + koff) * HIDDEN + m;
#pragma unroll
    for (int ct = 0; ct < 8; ++ct) {
      v2f b;
      b.x = bp[ct * 16];
      b.y = bp[ct * 16 + HIDDEN];
      acc0[ct] = __builtin_amdgcn_wmma_f32_16x16x4_f32(
          false, a0, false, b, (short)0, acc0[ct], false, false);
      acc1[ct] = __builtin_amdgcn_wmma_f32_16x16x4_f32(
          false, a1, false, b, (short)0, acc1[ct], false, false);
    }
  }

  // Stage 2 (per 16-row tile): ReLU + transpose via LDS, then GEMM2 with W1.
  auto stage2 = [&](v8f (&acc)[8], int t) {
    // C-layout -> row-major 16x128 tile in LDS (with ReLU).
#pragma unroll
    for (int ct = 0; ct < 8; ++ct)
#pragma unroll
      for (int r = 0; r < 8; ++r) {
        float v = acc[ct][r];
        hs[(rbase + r) * HIDDEN + ct * 16 + m] = v > 0.0f ? v : 0.0f;
      }
    // GEMM2: [16x128] @ [128x64], K = 128, 4 col tiles.
    v8f c2[4];
#pragma unroll
    for (int i = 0; i < 4; ++i)
#pragma unroll
      for (int j = 0; j < 8; ++j) c2[i][j] = 0.0f;

    for (int k = 0; k < HIDDEN; k += 4) {
      v2f a = *(const v2f*)(hs + m * HIDDEN + k + koff);  // ds_load_b64
      const float* bp = W1 + (size_t)(k + koff) * NCLS + m;
#pragma unroll
      for (int ct = 0; ct < 4; ++ct) {
        v2f b;
        b.x = bp[ct * 16];
        b.y = bp[ct * 16 + NCLS];
        c2[ct] = __builtin_amdgcn_wmma_f32_16x16x4_f32(
            false, a, false, b, (short)0, c2[ct], false, false);
      }
    }
    const int r0 = (tile0 + t) * 16;
#pragma unroll
    for (int ct = 0; ct < 4; ++ct)
#pragma unroll
      for (int r = 0; r < 8; ++r)
        H1[(size_t)(r0 + rbase + r) * NCLS + ct * 16 + m] = c2[ct][r];
  };

  stage2(acc0, 0);
  if (has2) stage2(acc1, 1);
}

// ---------------------------------------------------------------------------
// Zero both scatter-accumulation targets (float4 stores).
// ---------------------------------------------------------------------------
__global__ void __launch_bounds__(256) gcn_zero2_kernel(
    float4* __restrict__ a, float4* __restrict__ b, int n4)
{
  int i = blockIdx.x * 256 + (int)threadIdx.x;
  if (i < n4) {
    float4 z; z.x = 0.f; z.y = 0.f; z.z = 0.f; z.w = 0.f;
    a[i] = z;
    b[i] = z;
  }
}

// ---------------------------------------------------------------------------
// SpMM: z[dst] += val * x[src], 64-wide features.
// 16 threads/edge, 4 features each: float4 gather (L2-resident x),
// 4 hardware float atomics (global_atomic_add_f32).
// ---------------------------------------------------------------------------
__global__ void __launch_bounds__(256) gcn_spmm_kernel(
    const int* __restrict__ esrc, const int* __restrict__ edst,
    const float* __restrict__ eval, const float* __restrict__ xin,
    float* __restrict__ zout, int E)
{
  long long tid = (long long)blockIdx.x * 256 + threadIdx.x;
  int e = (int)(tid >> 4);
  if (e >= E) return;
  int f = ((int)tid & 15) << 2;

  int   s = esrc[e];
  int   d = edst[e];
  float v = eval[e];

  const float4 xv = *(const float4*)(xin + (size_t)s * NCLS + f);
  float* zp = zout + (size_t)d * NCLS + f;
  unsafeAtomicAdd(zp + 0, v * xv.x);
  unsafeAtomicAdd(zp + 1, v * xv.y);
  unsafeAtomicAdd(zp + 2, v * xv.z);
  unsafeAtomicAdd(zp + 3, v * xv.w);
}

// ---------------------------------------------------------------------------
extern "C" void kernel_launch(void* const* d_in, const int* in_sizes, int n_in,
                              void* d_out, int out_size, void* d_ws, size_t ws_size,
                              hipStream_t stream) {
  const float* X    = (const float*)d_in[0];  // [N, 512]
  const float* W0   = (const float*)d_in[1];  // [512, 128]
  const float* W1   = (const float*)d_in[2];  // [128, 64]
  const float* eVal = (const float*)d_in[3];  // [E]
  const int*   eSrc = (const int*)d_in[4];    // [E]
  const int*   eDst = (const int*)d_in[5];    // [E]
  float* out = (float*)d_out;                 // [N, 64]

  const int N = in_sizes[0] / IN_DIM;
  const int E = in_sizes[3];

  float* h1   = (float*)d_ws;                 // [N, 64]
  float* tbuf = h1 + (size_t)N * NCLS;        // [N, 64] ping buffer

  // 1) zero accumulation targets
  const int n4 = (N * NCLS) / 4;
  gcn_zero2_kernel<<<(n4 + 255) / 256, 256, 0, stream>>>(
      (float4*)tbuf, (float4*)out, n4);

  // 2) fused dense phase -> h1
  const int nTiles16    = (N + 15) / 16;
  const int wavesNeeded = (nTiles16 + 1) / 2;
  const int dblocks     = (wavesNeeded + 7) / 8;
  gcn_dense_kernel<<<dblocks, 256, 0, stream>>>(X, W0, W1, h1, nTiles16);

  // 3) two SpMM layers: h1 -> tbuf -> out
  const long long sThreads = (long long)E * 16;
  const int sblocks = (int)((sThreads + 255) / 256);
  gcn_spmm_kernel<<<sblocks, 256, 0, stream>>>(eSrc, eDst, eVal, h1, tbuf, E);
  gcn_spmm_kernel<<<sblocks, 256, 0, stream>>>(eSrc, eDst, eVal, tbuf, out, E);
}